// Quantize_30477087933017
// MI455X (gfx1250) — compile-verified
//
#include <hip/hip_runtime.h>
#include <stdint.h>

// ---------------- problem constants ----------------
#define DIMV   512
#define NEMB   4096
#define NROW   32768            // 32*32*32
#define EPSQ   1e-5f

// flat float offsets into d_out (tuple order: quantize, diff, embed_ind,
// new_embed, new_cluster_size, new_embed_avg)
#define O_Q    0
#define O_DIFF 16777216
#define O_IND  16777217
#define O_EMB  16809985
#define O_CS   18907137
#define O_AVG  18911233

typedef float        __attribute__((ext_vector_type(2))) v2f;
typedef float        __attribute__((ext_vector_type(8))) v8f;
typedef unsigned int __attribute__((ext_vector_type(4))) v4u;
typedef int          __attribute__((ext_vector_type(4))) v4i;
typedef int          __attribute__((ext_vector_type(8))) v8i;

__device__ __forceinline__ unsigned rfl(unsigned x) {
  return (unsigned)__builtin_amdgcn_readfirstlane((int)x);
}

// Wave32 sum-reduction through the matrix pipe:
// A(16x4 f32) holds each lane's partial in its K-slot, B = ones(4x16).
// D row-sums give (s_m + s_{m+16}); summing a lane's 8 D registers gives the
// half-wave sum, one shfl_xor(16) completes it. EXEC must be all-1s at call.
__device__ __forceinline__ float wave_sum_wmma(float s) {
  v2f a; a[0] = s;    a[1] = 0.0f;
  v2f b; b[0] = 1.0f; b[1] = 1.0f;
  v8f c = {};
  c = __builtin_amdgcn_wmma_f32_16x16x4_f32(false, a, false, b, (short)0, c,
                                            false, false);
  float t = 0.0f;
#pragma unroll
  for (int r = 0; r < 8; ++r) t += c[r];
  t += __shfl_xor(t, 16, 32);
  return t;
}

// ---------------- 1) init accumulators --------------------------------------
// new_embed_avg slot <- 99*embed_avg ; new_cluster_size slot <- 99*cluster_size
// (EMA trick: 0.99*x + 0.01*sum == 0.01*(99*x + sum)); zero diff and n.
__global__ void k_init(const float* __restrict__ ema,
                       const float* __restrict__ cs,
                       float* __restrict__ out, float* __restrict__ wsn) {
  int i = blockIdx.x * blockDim.x + threadIdx.x;
  if (i < DIMV * NEMB) out[O_AVG + i] = 99.0f * ema[i];
  if (i < NEMB)        out[O_CS + i]  = 99.0f * cs[i];
  if (i == 0) { out[O_DIFF] = 0.0f; *wsn = 0.0f; }
}

// ---------------- 2) transpose embed[512,4096] -> embedT[4096,512] ----------
// TDM gather pulls contiguous 2KB codebook rows, so we need embedT row-major.
__global__ void k_transpose(const float* __restrict__ embed,
                            float* __restrict__ embedT) {
  __shared__ float tile[32][33];
  int e0 = blockIdx.x * 32, d0 = blockIdx.y * 32;
  int tx = threadIdx.x, ty = threadIdx.y;
  tile[ty][tx] = embed[(size_t)(d0 + ty) * NEMB + e0 + tx];
  __syncthreads();
  embedT[(size_t)(e0 + ty) * DIMV + d0 + tx] = tile[tx][ty];
}

// ---------------- 3) scatter: segment-sum + histogram into accumulators -----
// One wave per row; 16.7M L2 f32 atomics total (memory-bound, no matmul form
// beats this without materializing a 512MB one-hot).
__global__ void k_scatter(const float* __restrict__ input,
                          const int* __restrict__ labels,
                          float* __restrict__ out) {
  int wid = threadIdx.x >> 5, lane = threadIdx.x & 31;
  int n = blockIdx.x * 8 + wid;
  int e = labels[n];
  if (lane == 0) atomicAdd(&out[O_CS + e], 1.0f);
  const float* row = input + (size_t)n * DIMV;
#pragma unroll
  for (int k = 0; k < DIMV / 32; ++k) {
    int d = k * 32 + lane;
    atomicAdd(&out[O_AVG + (size_t)d * NEMB + e], row[d]);
  }
}

// ---------------- 4) TDM gather + quantize store + MSE ----------------------
// Per wave: one gather-mode TENSOR_LOAD_TO_LDS pulls 16 label-indexed rows
// (16x512 f32 = 32KB) into LDS; TENSOR_STORE_FROM_LDS streams the tile to
// `quantize`; the wave computes the squared-error partial from LDS while the
// store drains, then reduces it through the WMMA pipe.
__global__ void k_gather(const float* __restrict__ embedT,
                         const int* __restrict__ labels,
                         const float* __restrict__ input,
                         float* __restrict__ out) {
  __shared__ float buf[2][16 * DIMV];      // 32KB per wave, 2 waves per block
  int wid = threadIdx.x >> 5, lane = threadIdx.x & 31;
  int rowbase = blockIdx.x * 32 + wid * 16;

  int lab = (lane < 16) ? labels[rowbase + lane] : 0;

  // pack 16 x 16-bit row indices into 8 uniform dwords (D# groups 2/3)
  unsigned idx[8];
#pragma unroll
  for (int i = 0; i < 8; ++i) {
    unsigned lo = (unsigned)__builtin_amdgcn_readlane(lab, 2 * i) & 0xFFFFu;
    unsigned hi = (unsigned)__builtin_amdgcn_readlane(lab, 2 * i + 1) & 0xFFFFu;
    idx[i] = lo | (hi << 16);
  }

  unsigned ldsOff = rfl((unsigned)(size_t)(&buf[wid][0]));

  // ---- load descriptor: gather 16 rows of 512 f32 from embedT ----
  unsigned long long gaL = (unsigned long long)(size_t)embedT;
  v4u g0 = { rfl(0x80000001u),              // count=1, gather_mode=1, idx16
             ldsOff,
             rfl((unsigned)gaL),
             rfl(((unsigned)(gaL >> 32) & 0x01FFFFFFu) | (2u << 30)) }; // type=2
  v8i g1 = { (int)(2u << 16),               // data_size = 4B
             (int)(512u << 16),             // tensor_dim0 = 512
             (int)(4096u << 16),            // tensor_dim1 = 4096 rows
             (int)(512u << 16),             // tile_dim0   = 512
             16,                            // tile_dim1   = 16 valid indices
             512,                           // tensor_dim0_stride = 512
             0, 0 };
  v4i g2 = { (int)idx[0], (int)idx[1], (int)idx[2], (int)idx[3] };
  v4i g3 = { (int)idx[4], (int)idx[5], (int)idx[6], (int)idx[7] };
  v8i pad = { 0, 0, 0, 0, 0, 0, 0, 0 };
  __builtin_amdgcn_tensor_load_to_lds(g0, g1, g2, g3, pad, 0);
  __builtin_amdgcn_s_wait_tensorcnt(0);

  // ---- store descriptor: contiguous 16x512 tile into quantize ----
  unsigned long long gaS =
      (unsigned long long)(size_t)(out + O_Q + (size_t)rowbase * DIMV);
  v4u s0 = { rfl(1u), ldsOff, rfl((unsigned)gaS),
             rfl(((unsigned)(gaS >> 32) & 0x01FFFFFFu) | (2u << 30)) };
  v8i s1 = { (int)(2u << 16),
             (int)(512u << 16),             // tensor_dim0 = 512
             (int)(16u << 16),              // tensor_dim1 = 16 rows
             (int)(512u << 16),             // tile_dim0   = 512
             16,                            // tile_dim1   = 16
             512,                           // tensor_dim0_stride = 512
             0, 0 };
  v4i z = { 0, 0, 0, 0 };
  __builtin_amdgcn_tensor_store_from_lds(s0, s1, z, z, pad, 0);

  // ---- MSE partial from LDS while the tensor store drains ----
  const float* xin = input + (size_t)rowbase * DIMV;
  float s = 0.0f;
#pragma unroll 4
  for (int j = 0; j < (16 * DIMV) / 32; ++j) {
    int t = j * 32 + lane;
    float d = buf[wid][t] - xin[t];
    s += d * d;
  }
  float tot = wave_sum_wmma(s);
  if (lane == 0)
    atomicAdd(&out[O_DIFF], tot * (1.0f / ((float)NROW * (float)DIMV)));
  if (lane < 16) out[O_IND + rowbase + lane] = (float)lab;
  // S_ENDPGM's implicit wait-idle covers the outstanding tensor store.
}

// ---------------- 5) cluster-size finalize + n reduction --------------------
__global__ void k_cs(float* __restrict__ out, float* __restrict__ wsn) {
  int i = blockIdx.x * 256 + threadIdx.x;
  float v = out[O_CS + i] * 0.01f;          // 0.99*cs + 0.01*hist
  out[O_CS + i] = v;
  float tot = wave_sum_wmma(v);
  if ((threadIdx.x & 31) == 0) atomicAdd(wsn, tot);
}

// ---------------- 6) embed_avg finalize + normalized codebook ---------------
__global__ void k_final(float* __restrict__ out, const float* __restrict__ wsn) {
  int i = blockIdx.x * 256 + threadIdx.x;   // 2,097,152 elems
  float ea = out[O_AVG + i] * 0.01f;        // 0.99*avg + 0.01*embed_sum
  out[O_AVG + i] = ea;
  int e = i & (NEMB - 1);
  float ncs = out[O_CS + e];
  float n = *wsn;
  float cs = (ncs + EPSQ) / (n + (float)NEMB * EPSQ) * n;
  out[O_EMB + i] = ea / cs;
}

extern "C" void kernel_launch(void* const* d_in, const int* in_sizes, int n_in,
                              void* d_out, int out_size, void* d_ws,
                              size_t ws_size, hipStream_t stream) {
  const float* input  = (const float*)d_in[0];
  const int*   labels = (const int*)d_in[1];
  const float* embed  = (const float*)d_in[2];
  const float* cs     = (const float*)d_in[3];
  const float* ema    = (const float*)d_in[4];
  float* out = (float*)d_out;
  float* embedT = (float*)d_ws;                                   // 8 MB
  float* wsn = (float*)((char*)d_ws + (size_t)8 * 1024 * 1024);   // n scalar

  k_init<<<8192, 256, 0, stream>>>(ema, cs, out, wsn);
  dim3 tb(32, 32), tg(NEMB / 32, DIMV / 32);
  k_transpose<<<tg, tb, 0, stream>>>(embed, embedT);
  k_scatter<<<NROW / 8, 256, 0, stream>>>(input, labels, out);
  k_gather<<<NROW / 32, 64, 0, stream>>>(embedT, labels, input, out);
  k_cs<<<NEMB / 256, 256, 0, stream>>>(out, wsn);
  k_final<<<(DIMV * NEMB) / 256, 256, 0, stream>>>(out, wsn);
}